// DigitalCapsule_15427522527270
// MI455X (gfx1250) — compile-verified
//
#include <hip/hip_runtime.h>

// ---------------------------------------------------------------------------
// Capsule dynamic routing, fused single kernel for MI455X (gfx1250).
//
// u_hat (302MB) is never materialized: x (4.7MB) + W (18.9MB) stay L2-resident
// (192MB L2) and u_hat tiles are recomputed with V_WMMA_F32_16X16X4_F32.
//   * s-pass:      s[b,d] = sum_{p,e} (c[b,p]*x[b,p,e]) * W[n,p,e,d]
//                  -> M=16b, K=8e per p, accumulated in the WMMA C operand.
//   * agreement:   a[b,p] = sum_e x[b,p,e] * (W_p . v^T)[e,b]
//                  -> g = W_p @ v^T as M=(2p x 8e), N=16b, K=16d WMMA GEMM;
//                     e-reduction register-local; per-row softmax max is
//                     tracked as a running register max inside the same loop
//                     (kills the separate max sweep over logits).
// One workgroup per (n, 16-wide b tile) = 256 blocks, 16 waves. Routing logits
// + softmax exps for all 1152 prev-capsules live in ~163KB LDS (<=320KB/WGP).
// HBM traffic ~= compulsory 24MB (~1us @ 23.3TB/s); rest is L2 + LDS + WMMA.
// ---------------------------------------------------------------------------

#define B_      128
#define N_      32
#define P_      1152
#define PD_     8
#define D_      16
#define WAVES   16
#define THREADS (WAVES * 32)
#define EPS_    1e-8f

typedef __attribute__((ext_vector_type(2))) float v2f;
typedef __attribute__((ext_vector_type(8))) float v8f;

__device__ __forceinline__ v8f wmma4(v2f a, v2f b, v8f c) {
  return __builtin_amdgcn_wmma_f32_16x16x4_f32(false, a, false, b, (short)0, c,
                                               false, false);
}

// Agreement pass, specialized on accumulate-vs-assign so the inner loop has no
// per-iteration branch. Returns this lane's running max over the logits it
// wrote (lane (half,l16) only ever writes rows b == l16).
template <bool ACC>
__device__ __forceinline__ float agree_loop(const float* __restrict__ x,
                                            const float* __restrict__ Wn,
                                            float* __restrict__ logits,
                                            const float* __restrict__ vsh,
                                            int b0, int wv, int half, int l16,
                                            int k2, int ap, int ae) {
  v2f bv[4];
  #pragma unroll
  for (int j = 0; j < 4; ++j) {                   // B[k=d][n=b] = v[b][d]
    bv[j].x = vsh[l16 * 16 + 4 * j + k2];
    bv[j].y = vsh[l16 * 16 + 4 * j + k2 + 1];
  }
  float lmax = -3.0e38f;
  for (int pp = wv * 2; pp < P_; pp += WAVES * 2) {            // p-pair
    const float* wa = Wn + ((size_t)(pp + ap) * PD_ + ae) * D_ + k2;
    v8f g = {};
    #pragma unroll
    for (int j = 0; j < 4; ++j) {                 // A[m=(p,e)][k=d], b64 loads
      v2f a; a.x = wa[4 * j]; a.y = wa[4 * j + 1];
      g = wmma4(a, bv[j], g);
    }
    // lane (half,l16) holds g[p_{half}, e=r, b=l16] in g[r]
    const float* xq = x + ((size_t)(b0 + l16) * P_ + (pp + half)) * PD_;
    const float4 x0 = *(const float4*)(xq);
    const float4 x1 = *(const float4*)(xq + 4);
    float a = x0.x * g[0] + x0.y * g[1] + x0.z * g[2] + x0.w * g[3] +
              x1.x * g[4] + x1.y * g[5] + x1.z * g[6] + x1.w * g[7];
    const int idx = (pp + half) * 16 + l16;       // 32 lanes -> 32 banks
    if (ACC) a += logits[idx];
    logits[idx] = a;
    lmax = fmaxf(lmax, a);
  }
  return lmax;
}

__global__ __launch_bounds__(THREADS, 1)
void caps_routing_kernel(const float* __restrict__ x,
                         const float* __restrict__ W,
                         float* __restrict__ out) {
  extern __shared__ float smem[];
  float* logits = smem;               // [P][16]   routing logits (p-major)
  float* cexp   = logits + P_ * 16;   // [P][16]   unnormalized softmax exp
  float* spart  = cexp   + P_ * 16;   // [WAVES][16][16] per-wave s partials
  float* vsh    = spart  + WAVES*256; // [16][16]  v
  float* redtmp = vsh    + 256;       // [32][16]  reduction scratch
  float* rmax   = redtmp + 512;       // [16]
  float* rinv   = rmax   + 16;        // [16]

  const int tid  = threadIdx.x;
  const int wv   = tid >> 5;          // wave 0..15
  const int lid  = tid & 31;
  const int half = lid >> 4;
  const int l16  = lid & 15;
  const int k2   = half * 2;          // K sub-block select (ISA A/B lane split)

  const int n  = blockIdx.x >> 3;
  const int b0 = (blockIdx.x & 7) << 4;

  const float* xrow = x + (size_t)(b0 + l16) * P_ * PD_;     // s-pass A row b=l16
  const float* Wn   = W + (size_t)n * P_ * PD_ * D_;
  const float* Wnd  = Wn + l16;                              // s-pass B col d=l16
  const int ap = l16 >> 3;            // agreement: p-of-pair for A row m=l16
  const int ae = l16 & 7;             // agreement: e for A row m=l16

  const int bidx = tid >> 4;          // reduction-phase b (tid<256)
  const int didx = tid & 15;

  for (int it = 0; it < 3; ++it) {
    if (it > 0) {
      // ---- agreement + fused running row-max ----
      float lmax = (it == 1)
          ? agree_loop<false>(x, Wn, logits, vsh, b0, wv, half, l16, k2, ap, ae)
          : agree_loop<true >(x, Wn, logits, vsh, b0, wv, half, l16, k2, ap, ae);
      lmax = fmaxf(lmax, __shfl_xor(lmax, 16, 32));   // merge the two halves
      if (lid < 16) redtmp[wv * 16 + l16] = lmax;     // 16 waves x 16 b
      __syncthreads();
      if (tid < 16) {
        float mm = -3.0e38f;
        #pragma unroll
        for (int w = 0; w < WAVES; ++w) mm = fmaxf(mm, redtmp[w * 16 + tid]);
        rmax[tid] = mm;
      }
      __syncthreads();
      // ---- exp + sum sweep (the only full pass over logits) ----
      {
        const int pb = tid & 15, pg = tid >> 4;       // pg 0..31
        const float mb = rmax[pb];
        float sum = 0.f;
        for (int p = pg; p < P_; p += 32) {
          const float e = __expf(logits[p * 16 + pb] - mb);
          cexp[p * 16 + pb] = e;
          sum += e;
        }
        redtmp[pg * 16 + pb] = sum;
        __syncthreads();
        if (tid < 16) {
          float ss = 0.f;
          #pragma unroll
          for (int g = 0; g < 32; ++g) ss += redtmp[g * 16 + tid];
          rinv[tid] = 1.f / ss;
        }
        __syncthreads();
      }
    }

    // ---- s pass: accumulate s[b,d] in the WMMA C operand across p ----
    v8f sa = {};
    if (it == 0) {                                // softmax(0) == uniform
      for (int p = wv; p < P_; p += WAVES) {
        const float* xp = xrow + (size_t)p * PD_;
        v2f a0, a1, bm0, bm1;
        a0.x = xp[k2];     a0.y = xp[k2 + 1];
        a1.x = xp[k2 + 4]; a1.y = xp[k2 + 5];
        const float* wp = Wnd + (size_t)p * (PD_ * D_);
        bm0.x = wp[(k2)     * D_]; bm0.y = wp[(k2 + 1) * D_];
        bm1.x = wp[(k2 + 4) * D_]; bm1.y = wp[(k2 + 5) * D_];
        sa = wmma4(a0, bm0, sa);
        sa = wmma4(a1, bm1, sa);
      }
    } else {                                      // fold c into A: y = c*x
      for (int p = wv; p < P_; p += WAVES) {
        const float cc = cexp[p * 16 + l16];      // c[b=l16, p], broadcast read
        const float* xp = xrow + (size_t)p * PD_;
        v2f a0, a1, bm0, bm1;
        a0.x = xp[k2]     * cc; a0.y = xp[k2 + 1] * cc;
        a1.x = xp[k2 + 4] * cc; a1.y = xp[k2 + 5] * cc;
        const float* wp = Wnd + (size_t)p * (PD_ * D_);
        bm0.x = wp[(k2)     * D_]; bm0.y = wp[(k2 + 1) * D_];
        bm1.x = wp[(k2 + 4) * D_]; bm1.y = wp[(k2 + 5) * D_];
        sa = wmma4(a0, bm0, sa);
        sa = wmma4(a1, bm1, sa);
      }
    }
    #pragma unroll
    for (int r = 0; r < 8; ++r)                   // C layout: b=r+half*8, d=l16
      spart[wv * 256 + (r + half * 8) * 16 + l16] = sa[r];
    __syncthreads();

    // ---- cross-wave reduce, normalize, squash (first 256 threads) ----
    if (tid < 256) {
      float s = 0.f;
      #pragma unroll
      for (int w = 0; w < WAVES; ++w) s += spart[w * 256 + tid];
      s *= (it == 0) ? (1.f / (float)P_) : rinv[bidx];
      float sq = s * s;                           // ||s||^2 over d (16-lane grp)
      sq += __shfl_xor(sq, 1, 32);
      sq += __shfl_xor(sq, 2, 32);
      sq += __shfl_xor(sq, 4, 32);
      sq += __shfl_xor(sq, 8, 32);
      const float scale = (sq / (1.f + sq)) * rsqrtf(sq + EPS_);
      const float vv = scale * s;
      if (it == 2)
        out[((size_t)(b0 + bidx) * N_ + n) * D_ + didx] = vv;  // [B,N,D]
      else
        vsh[tid] = vv;
    }
    __syncthreads();
  }
}

extern "C" void kernel_launch(void* const* d_in, const int* in_sizes, int n_in,
                              void* d_out, int out_size, void* d_ws, size_t ws_size,
                              hipStream_t stream) {
  (void)in_sizes; (void)n_in; (void)out_size; (void)d_ws; (void)ws_size;
  const float* x = (const float*)d_in[0];   // [128,1152,8]
  const float* W = (const float*)d_in[1];   // [32,1152,8,16]
  float* out = (float*)d_out;               // [128,32,16]

  const size_t smem_bytes =
      (size_t)(2 * P_ * 16 + WAVES * 256 + 256 + 512 + 16 + 16) * sizeof(float);

  (void)hipFuncSetAttribute((const void*)caps_routing_kernel,
                            hipFuncAttributeMaxDynamicSharedMemorySize,
                            (int)smem_bytes);

  caps_routing_kernel<<<dim3(N_ * (B_ / 16)), dim3(THREADS), smem_bytes,
                        stream>>>(x, W, out);
}